// SIFTNet_28037546508492
// MI455X (gfx1250) — compile-verified
//
#include <hip/hip_runtime.h>
#include <hip/hip_bf16.h>

typedef __attribute__((ext_vector_type(2))) float v2f;
typedef __attribute__((ext_vector_type(8))) float v8f;

#define AS1 __attribute__((address_space(1)))
#define AS3 __attribute__((address_space(3)))

#define TS 32  // LDS tile row stride (floats)

#if __has_builtin(__builtin_amdgcn_global_load_async_to_lds_b32)
#define HAVE_ASYNC_LDS 1
#else
#define HAVE_ASYNC_LDS 0
#endif

// One workgroup (256 threads = 8 wave32) per image.
__launch_bounds__(256)
__global__ void sift_fused(const float* __restrict__ x,        // (2048,1,64,64)
                           const float* __restrict__ w_sobel,  // (2,1,3,3)
                           const float* __restrict__ w_orient, // (10,2,1,1)
                           const float* __restrict__ w_acc,    // (8,1,4,4)
                           float* __restrict__ out)            // (2048,7,128)
{
    __shared__ float T[30 * TS];     // input tile, origin at image coord (-1,-1)
    __shared__ float Lnorm[784];     // per-pixel gradient magnitude, 28x28
    __shared__ int   Lbin[784];      // per-pixel argmax orientation bin
    __shared__ float Acc[169 * 8];   // 13x13 even-grid acc values, 8 channels
    __shared__ float Wacc[128];      // depthwise 4x4 taps

    const int tid  = threadIdx.x;
    const int lane = tid & 31;
    const int wid  = tid >> 5;
    const int b    = blockIdx.x;
    const float* __restrict__ img = x + (size_t)b * 4096;

    // Prefetch the live region of the next image (gfx1250 global_prefetch_b8).
    if (b + 1 < (int)gridDim.x && tid < 29)
        __builtin_prefetch(x + (size_t)(b + 1) * 4096 + tid * 64, 0, 1);

    if (tid < 128) Wacc[tid] = w_acc[tid];

#if HAVE_ASYNC_LDS
    // Zero-fill the pad row (ty=0) and pad col (tx=0) with plain LDS stores.
    if (tid < 30) { T[tid] = 0.f; T[tid * TS] = 0.f; }
    // Stage the 29x29 live window with CDNA5 async global->LDS DMA (ASYNCcnt).
    for (int i = tid; i < 29 * 29; i += 256) {
        const int ty = i / 29 + 1;
        const int tx = i - (ty - 1) * 29 + 1;
        const float* gsrc = img + (ty - 1) * 64 + (tx - 1);
        __builtin_amdgcn_global_load_async_to_lds_b32(
            (AS1 int*)(unsigned long long)gsrc,
            (AS3 int*)(unsigned int)(unsigned long long)&T[ty * TS + tx],
            0, 0);
    }
#if __has_builtin(__builtin_amdgcn_s_wait_asynccnt)
    __builtin_amdgcn_s_wait_asynccnt(0);
#else
    asm volatile("s_wait_asynccnt 0x0" ::: "memory");
#endif
#else
    // Fallback: stage through VGPRs.
    for (int i = tid; i < 30 * TS; i += 256) {
        const int ty = i >> 5, tx = i & 31;
        if (tx < 30) {
            const int iy = ty - 1, ix = tx - 1;
            T[ty * TS + tx] = (iy >= 0 && ix >= 0) ? img[iy * 64 + ix] : 0.f;
        }
    }
#endif

    // Sobel taps from the actual input tensor (uniform -> scalar loads).
    float kx[9], ky[9];
#pragma unroll
    for (int i = 0; i < 9; ++i) { kx[i] = w_sobel[i]; ky[i] = w_sobel[9 + i]; }

    // A-matrix (16x4 f32): rows 0..7 = orientation weights, rest zero.
    // Layout: v0 = K0 (lanes 0-15) / K2 (lanes 16-31); v1 = K1 / K3.
    const int  ci  = lane & 7;
    const bool isA = lane < 8;
    v2f av;
    av.x = isA ? w_orient[2 * ci]     : 0.f;
    av.y = isA ? w_orient[2 * ci + 1] : 0.f;
    const float wi00 = w_orient[16], wi01 = w_orient[17];  // im_grads rows
    const float wi10 = w_orient[18], wi11 = w_orient[19];

    __syncthreads();

    // 784 = 49 groups of 16 pixels; each wave handles groups wid, wid+8, ...
    const bool lo = lane < 16;
    for (int g = wid; g < 49; g += 8) {
        const int p  = g * 16 + (lane & 15);   // lanes n and n+16 mirror pixel n
        const int py = p / 28;
        const int px = p - py * 28;

        float gxv = 0.f, gyv = 0.f;
#pragma unroll
        for (int kk = 0; kk < 9; ++kk) {
            const float tv = T[(py + kk / 3) * TS + (px + kk % 3)];
            gxv = fmaf(kx[kk], tv, gxv);
            gyv = fmaf(ky[kk], tv, gyv);
        }

        // B-matrix (4x16 f32): row0 = gx, row1 = gy, rows 2-3 zero.
        v2f bv;
        bv.x = lo ? gxv : 0.f;
        bv.y = lo ? gyv : 0.f;
        v8f acc = {0.f, 0.f, 0.f, 0.f, 0.f, 0.f, 0.f, 0.f};
        // D[c][n] = cos_c*gx_n + sin_c*gy_n ; lane n<16 holds channels 0..7.
        acc = __builtin_amdgcn_wmma_f32_16x16x4_f32(
                  false, av, false, bv, (short)0, acc, false, false);

        // In-lane argmax over the 8 orientation cosines (first-max on ties).
        float best = acc[0];
        int bin = 0;
#pragma unroll
        for (int c = 1; c < 8; ++c)
            if (acc[c] > best) { best = acc[c]; bin = c; }

        const float i0  = wi00 * gxv + wi01 * gyv;
        const float i1  = wi10 * gxv + wi11 * gyv;
        const float nrm = sqrtf(i0 * i0 + i1 * i1);
        if (lo) { Lbin[p] = bin; Lnorm[p] = nrm; }   // divergence only after WMMA
    }
    __syncthreads();

    // acc(py,px,c) on the 13x13 even grid: 4x4 window over sparse one-hot hist.
    if (tid < 169) {
        const int gy = tid / 13;
        const int gxp = tid - gy * 13;
        const int r0 = 2 * gy, c0 = 2 * gxp;
        float a8[8] = {0.f, 0.f, 0.f, 0.f, 0.f, 0.f, 0.f, 0.f};
#pragma unroll
        for (int i = 0; i < 4; ++i) {
#pragma unroll
            for (int j = 0; j < 4; ++j) {
                const int pp = (r0 + i) * 28 + (c0 + j);
                const int bn = Lbin[pp];
                const float contrib = Wacc[bn * 16 + i * 4 + j] * Lnorm[pp];
#pragma unroll
                for (int c = 0; c < 8; ++c)
                    a8[c] += (bn == c) ? contrib : 0.f;
            }
        }
#pragma unroll
        for (int c = 0; c < 8; ++c) Acc[tid * 8 + c] = a8[c];
    }
    __syncthreads();

    // 7 scales -> 7 waves; gather 16 points x 8 ch, L2-normalize, v^0.9 on >0.
    if (tid < 224) {
        const int s = wid;             // scale 0..6
        float vals[4];
        float ss = 0.f;
#pragma unroll
        for (int j = 0; j < 4; ++j) {
            const int k    = lane + 32 * j;      // feature index 0..127
            const int pidx = k >> 3, cch = k & 7;
            const int yi   = pidx >> 2, xi = pidx & 3;
            const int gpos = (s + 2 * yi) * 13 + (s + 2 * xi);
            const float v  = Acc[gpos * 8 + cch];
            vals[j] = v;
            ss += v * v;
        }
#pragma unroll
        for (int off = 16; off > 0; off >>= 1)
            ss += __shfl_xor(ss, off, 32);
        const float rinv = (ss > 0.f) ? rsqrtf(ss) : 0.f;
        float* op = out + (size_t)b * 896 + (size_t)s * 128;
#pragma unroll
        for (int j = 0; j < 4; ++j) {
            const int k   = lane + 32 * j;
            const float v = vals[j] * rinv;
            op[k] = (v > 0.f) ? exp2f(0.9f * log2f(v)) : 0.f;
        }
    }
}

extern "C" void kernel_launch(void* const* d_in, const int* in_sizes, int n_in,
                              void* d_out, int out_size, void* d_ws, size_t ws_size,
                              hipStream_t stream) {
    const float* x        = (const float*)d_in[0];
    const float* w_sobel  = (const float*)d_in[1];
    const float* w_orient = (const float*)d_in[2];
    const float* w_acc    = (const float*)d_in[3];
    float* out = (float*)d_out;
    const int B = in_sizes[0] / 4096;   // 2048 images of 64*64
    hipLaunchKernelGGL(sift_fused, dim3(B), dim3(256), 0, stream,
                       x, w_sobel, w_orient, w_acc, out);
}